// LinearAttentionCell_89764816486872
// MI455X (gfx1250) — compile-verified
//
#include <hip/hip_runtime.h>

// Reference (T=1) reduces algebraically to: out[64,2048] = x[64,2048] @ Wv[2048,2048]^T
// (memory = k (x) v outer product; mem_read = (q.k) * v; scale = q.k; out = v.)
// Single f32 GEMM via V_WMMA_F32_16X16X4_F32 (full f32 precision, matrix pipe).

typedef float v2f __attribute__((ext_vector_type(2)));
typedef float v8f __attribute__((ext_vector_type(8)));

#define K_DIM 2048
#define N_DIM 2048

__global__ __launch_bounds__(256) void v_gemm_wmma_f32(
    const float* __restrict__ x,   // [64, 2048] row-major (K contiguous)
    const float* __restrict__ Wv,  // [2048, 2048] row-major (K contiguous) -> B column-major
    float* __restrict__ out)       // [64, 2048]
{
    const int lane = threadIdx.x & 31;
    const int wave = threadIdx.x >> 5;
    const int task = blockIdx.x * 8 + wave;   // 0..255
    const int mt   = task & 3;                // M-tile: rows mt*16 .. mt*16+15
    const int n0   = (task >> 2) * 32;        // 32-wide N block (2 tiles of 16)

    // WMMA f32 16x16x4 fragment mapping (ISA 7.12.2):
    //   A: lane l -> M = l&15, VGPR r holds K = (l>>4)*2 + r
    //   B: lane l -> N = l&15, VGPR r holds K = (l>>4)*2 + r
    const int m   = lane & 15;
    const int khi = (lane >> 4) * 2;          // 0 or 2

    const float* xrow  = x  + (size_t)(mt * 16 + m) * K_DIM + khi;
    const float* wrow0 = Wv + (size_t)(n0 +  0 + m) * K_DIM + khi;
    const float* wrow1 = Wv + (size_t)(n0 + 16 + m) * K_DIM + khi;

    // 4 independent accumulation chains: 2 N-tiles x 2 K-phases
    v8f acc00 = {}, acc01 = {}, acc10 = {}, acc11 = {};

    #pragma unroll 4
    for (int k = 0; k < K_DIM; k += 8) {
        v2f a0  = *(const v2f*)(xrow  + k);
        v2f a1  = *(const v2f*)(xrow  + k + 4);
        v2f b00 = *(const v2f*)(wrow0 + k);
        v2f b01 = *(const v2f*)(wrow0 + k + 4);
        v2f b10 = *(const v2f*)(wrow1 + k);
        v2f b11 = *(const v2f*)(wrow1 + k + 4);
        // 8 args: (neg_a, A, neg_b, B, c_mod, C, reuse_a, reuse_b)
        acc00 = __builtin_amdgcn_wmma_f32_16x16x4_f32(false, a0, false, b00, (short)0, acc00, false, false);
        acc10 = __builtin_amdgcn_wmma_f32_16x16x4_f32(false, a0, false, b10, (short)0, acc10, false, false);
        acc01 = __builtin_amdgcn_wmma_f32_16x16x4_f32(false, a1, false, b01, (short)0, acc01, false, false);
        acc11 = __builtin_amdgcn_wmma_f32_16x16x4_f32(false, a1, false, b11, (short)0, acc11, false, false);
    }

    v8f accA = acc00 + acc01;   // N-tile 0
    v8f accB = acc10 + acc11;   // N-tile 1

    // C/D layout: VGPR r, lane l -> M = r + 8*(l>>4), N = l&15
    const int mrow = mt * 16 + 8 * (lane >> 4);
    const int ncol = n0 + (lane & 15);
    #pragma unroll
    for (int r = 0; r < 8; ++r) {
        out[(size_t)(mrow + r) * N_DIM + ncol +  0] = accA[r];
        out[(size_t)(mrow + r) * N_DIM + ncol + 16] = accB[r];
    }
}

extern "C" void kernel_launch(void* const* d_in, const int* in_sizes, int n_in,
                              void* d_out, int out_size, void* d_ws, size_t ws_size,
                              hipStream_t stream) {
    // setup_inputs order: x, Wq, bq, Wk, bk, Wv, pos
    const float* x  = (const float*)d_in[0];
    const float* Wv = (const float*)d_in[5];
    float* out = (float*)d_out;

    // 256 wave-tasks (4 M-tiles x 64 N-blocks of 32 cols), 8 waves per 256-thread block
    v_gemm_wmma_f32<<<32, 256, 0, stream>>>(x, Wv, out);
}